// FAPELoss_77799037600139
// MI455X (gfx1250) — compile-verified
//
#include <hip/hip_runtime.h>
#include <hip/hip_bf16.h>

typedef __attribute__((ext_vector_type(2))) float v2f;
typedef __attribute__((ext_vector_type(8))) float v8f;

#define SEQ_L 2048
#define NCOLS (SEQ_L * 3)          // 6144 residue-atom columns
#define CLAMP_D 10.0f
#define LOSS_UNIT 10.0f
#define FAPE_EPS 1e-8f

// ---------------------------------------------------------------------------
// Main kernel: each wave owns 4 frames. A-matrix (rotations) is loop-invariant;
// B-matrix streams 16 residue-atom columns per iteration. One f32 WMMA per
// (pred, true) per tile. D-row packing keeps a frame's 3 components in one lane.
// ---------------------------------------------------------------------------
__global__ __launch_bounds__(256) void fape_main_kernel(
    const float* __restrict__ pc,   // pred_coords      (L,3,3)
    const float* __restrict__ tc,   // true_coords      (L,3,3)
    const float* __restrict__ pr,   // pred_rotation    (L,3,3)
    const float* __restrict__ pt,   // pred_translation (L,3)
    const float* __restrict__ tr,   // true_rotation    (L,3,3)
    const float* __restrict__ tt,   // true_translation (L,3)
    const int*   __restrict__ mask, // (L,)
    float* __restrict__ blockPartials)
{
    const int tid  = threadIdx.x;
    const int wave = tid >> 5;          // 0..7
    const int lane = tid & 31;
    const int half = lane >> 4;         // A/B: K-half (c in {0,1} vs {2,3}); D: M-half
    const int l15  = lane & 15;

    const int frameBase = (blockIdx.x * 8 + wave) * 4;   // 4 frames per wave

    // ---- Build loop-invariant A matrices (16x4, f32): rows -> (frame, d)
    //   M in [0,6)  : frame frameBase   + M/3,     d = M%3      (D half 0)
    //   M in [8,14) : frame frameBase+2 + (M-8)/3, d = (M-8)%3  (D half 1)
    //   rows 6,7,14,15 and K=3 are zero padding.
    // Lane layout of A (32-bit, 2 VGPRs): lanes 0-15 hold K=0,1 of row M=lane;
    // lanes 16-31 hold K=2,3 of row M=lane-16.
    const int M    = l15;
    const int sub  = (M < 8) ? M : (M - 8);
    const bool rowValid = (sub < 6);
    const int fOff = ((M < 8) ? 0 : 2) + sub / 3;
    const int dIdx = sub % 3;
    const int frameA = frameBase + fOff;

    v2f aP = {0.0f, 0.0f};
    v2f aT = {0.0f, 0.0f};
    if (rowValid) {
        const int c0 = half * 2;                 // 0 or 2
        const float* rpRow = pr + frameA * 9 + dIdx * 3;
        const float* rtRow = tr + frameA * 9 + dIdx * 3;
        aP.x = rpRow[c0];
        aT.x = rtRow[c0];
        if (half == 0) {                         // c=1; c=3 stays zero pad
            aP.y = rpRow[1];
            aT.y = rtRow[1];
        }
    }

    // ---- Per-lane frame pair for the D layout: lanes of half h own frames
    // fA = frameBase + 2h (D vgprs 0..2) and fB = fA+1 (D vgprs 3..5).
    const int fA = frameBase + 2 * half;
    const int fB = fA + 1;

    // delta[d] = (R_p t_p)[d] - (R_t t_t)[d], folded translation constant.
    float delA0 = 0.f, delA1 = 0.f, delA2 = 0.f;
    float delB0 = 0.f, delB1 = 0.f, delB2 = 0.f;
    {
        float dA[3], dB[3];
        #pragma unroll
        for (int d = 0; d < 3; ++d) {
            float spA = 0.f, stA = 0.f, spB = 0.f, stB = 0.f;
            #pragma unroll
            for (int c = 0; c < 3; ++c) {
                spA += pt[fA * 3 + c] * pr[fA * 9 + d * 3 + c];
                stA += tt[fA * 3 + c] * tr[fA * 9 + d * 3 + c];
                spB += pt[fB * 3 + c] * pr[fB * 9 + d * 3 + c];
                stB += tt[fB * 3 + c] * tr[fB * 9 + d * 3 + c];
            }
            dA[d] = spA - stA;
            dB[d] = spB - stB;
        }
        delA0 = dA[0]; delA1 = dA[1]; delA2 = dA[2];
        delB0 = dB[0]; delB1 = dB[1]; delB2 = dB[2];
    }
    const float mFA = (float)mask[fA];
    const float mFB = (float)mask[fB];

    // ---- Sweep all 6144 columns, 16 per WMMA tile.
    float accA = 0.0f, accB = 0.0f;
    const v8f zeroC = {};
    for (int colBase = 0; colBase < NCOLS; colBase += 16) {
        const int n = colBase + l15;   // this lane's column (B lanes: N=lane&15)

        // B (4x16, f32, 2 VGPRs): lanes 0-15 hold K=0,1 of col N=lane;
        // lanes 16-31 hold K=2,3(pad) of col N=lane-16.  B[c][n] = coords[n*3+c].
        v2f bP, bT;
        const float* pcn = pc + n * 3;
        const float* tcn = tc + n * 3;
        if (half == 0) {
            bP.x = pcn[0]; bP.y = pcn[1];
            bT.x = tcn[0]; bT.y = tcn[1];
        } else {
            bP.x = pcn[2]; bP.y = 0.0f;
            bT.x = tcn[2]; bT.y = 0.0f;
        }

        v8f Dp = __builtin_amdgcn_wmma_f32_16x16x4_f32(
            false, aP, false, bP, (short)0, zeroC, false, false);
        v8f Dt = __builtin_amdgcn_wmma_f32_16x16x4_f32(
            false, aT, false, bT, (short)0, zeroC, false, false);

        const float ml = (float)mask[n / 3];

        // frame fA: rows (half*8)+0..2  -> D vgprs 0..2 of this lane
        float dx = Dp[0] - Dt[0] - delA0;
        float dy = Dp[1] - Dt[1] - delA1;
        float dz = Dp[2] - Dt[2] - delA2;
        float d2 = dx * dx + dy * dy + dz * dz + FAPE_EPS;
        accA += fminf(__builtin_sqrtf(d2), CLAMP_D) * ml;

        // frame fB: rows (half*8)+3..5 -> D vgprs 3..5
        dx = Dp[3] - Dt[3] - delB0;
        dy = Dp[4] - Dt[4] - delB1;
        dz = Dp[5] - Dt[5] - delB2;
        d2 = dx * dx + dy * dy + dz * dz + FAPE_EPS;
        accB += fminf(__builtin_sqrtf(d2), CLAMP_D) * ml;
    }

    float acc = accA * mFA + accB * mFB;

    // ---- Deterministic block reduction.
    __shared__ float red[256];
    red[tid] = acc;
    __syncthreads();
    #pragma unroll
    for (int s = 128; s > 0; s >>= 1) {
        if (tid < s) red[tid] += red[tid + s];
        __syncthreads();
    }
    if (tid == 0) blockPartials[blockIdx.x] = red[0];
}

// ---------------------------------------------------------------------------
// Finalize: fixed-order reduce of 64 block partials + mask sum, then scale.
// ---------------------------------------------------------------------------
__global__ __launch_bounds__(256) void fape_finalize_kernel(
    const float* __restrict__ partials, int nPartials,
    const int* __restrict__ mask, float* __restrict__ out)
{
    __shared__ float sNum[256];
    __shared__ float sMsk[256];
    const int tid = threadIdx.x;

    float a = (tid < nPartials) ? partials[tid] : 0.0f;
    float msum = 0.0f;
    for (int i = tid; i < SEQ_L; i += 256) msum += (float)mask[i];
    sNum[tid] = a;
    sMsk[tid] = msum;
    __syncthreads();
    #pragma unroll
    for (int s = 128; s > 0; s >>= 1) {
        if (tid < s) {
            sNum[tid] += sNum[tid + s];
            sMsk[tid] += sMsk[tid + s];
        }
        __syncthreads();
    }
    if (tid == 0) {
        const float numer = sNum[0];
        const float m     = sMsk[0];
        const float denom = m * m * 3.0f + FAPE_EPS;
        out[0] = numer / denom / LOSS_UNIT;
    }
}

extern "C" void kernel_launch(void* const* d_in, const int* in_sizes, int n_in,
                              void* d_out, int out_size, void* d_ws, size_t ws_size,
                              hipStream_t stream) {
    const float* pc = (const float*)d_in[0];   // pred_coords      (L,3,3)
    const float* tc = (const float*)d_in[1];   // true_coords      (L,3,3)
    const float* pr = (const float*)d_in[2];   // pred_rotation    (L,3,3)
    const float* pt = (const float*)d_in[3];   // pred_translation (L,3)
    const float* tr = (const float*)d_in[4];   // true_rotation    (L,3,3)
    const float* tt = (const float*)d_in[5];   // true_translation (L,3)
    const int*   mk = (const int*)d_in[6];     // mask (L,)
    float* out = (float*)d_out;

    float* partials = (float*)d_ws;            // 64 floats of scratch

    const int nBlocks = SEQ_L / 32;            // 2048 frames / (8 waves * 4 frames) = 64
    fape_main_kernel<<<nBlocks, 256, 0, stream>>>(pc, tc, pr, pt, tr, tt, mk, partials);
    fape_finalize_kernel<<<1, 256, 0, stream>>>(partials, nBlocks, mk, out);
}